// Net_50448685859415
// MI455X (gfx1250) — compile-verified
//
#include <hip/hip_runtime.h>

typedef __attribute__((ext_vector_type(2))) float v2f;
typedef __attribute__((ext_vector_type(8))) float v8f;

#define TPB 256
#define WAVES_PER_BLOCK 8  // 256 threads / wave32

// ---------------------------------------------------------------------------
// deg[n] = 1 (self loop); then atomic count over edge destinations; then rsqrt
// ---------------------------------------------------------------------------
__global__ __launch_bounds__(TPB) void k_fill_deg(float* deg, int n) {
    int i = blockIdx.x * TPB + threadIdx.x;
    if (i < n) deg[i] = 1.0f;  // self-loop contributes 1 to every node's degree
}

__global__ __launch_bounds__(TPB) void k_deg_edges(const int* __restrict__ dst,
                                                   float* deg, int e) {
    int i = blockIdx.x * TPB + threadIdx.x;
    if (i < e) atomicAdd(&deg[dst[i]], 1.0f);
}

__global__ __launch_bounds__(TPB) void k_dis(float* deg, int n) {
    int i = blockIdx.x * TPB + threadIdx.x;
    if (i < n) {
        float d = deg[i];
        deg[i] = (d > 0.0f) ? rsqrtf(fmaxf(d, 1e-12f)) : 0.0f;  // in place
    }
}

// ---------------------------------------------------------------------------
// out[N x 16] = op(A[xidx] [N x 16]) @ W[16 x 16] via V_WMMA_F32_16X16X4_F32.
// One wave per 16-row tile; K=16 done as 4 chained k=4 WMMAs.
//   A lane layout (16x4 f32): lane<16 -> K = kc+0,kc+1 ; lane>=16 -> K = kc+2,kc+3
//   B lane layout (4x16 f32): VGPR j = row (kc + j + 2*(lane>>4)), col lane&15
//   D layout (16x16 f32):     VGPR v = row (v + 8*(lane>>4)),      col lane&15
// GATHER/RELU are compile-time so each launch is fully specialized (no
// v_cndmask selects in the inner loop, no dead max ops).
// ---------------------------------------------------------------------------
template <bool GATHER, bool RELU>
__global__ __launch_bounds__(TPB)
void k_gemm16_wmma(const float* __restrict__ A, const int* __restrict__ xidx,
                   const float* __restrict__ W, float* __restrict__ out,
                   int n) {
    const int lane = threadIdx.x & 31;
    const int wave = threadIdx.x >> 5;
    const int tile = blockIdx.x * WAVES_PER_BLOCK + wave;
    const int ntiles = (n + 15) >> 4;
    if (tile >= ntiles) return;           // wave-uniform: EXEC stays all-1s

    const int col = lane & 15;            // row-in-tile for A, column for B/D
    const int g   = lane >> 4;            // half-wave group

    int row  = tile * 16 + col;
    int lrow = (row < n) ? row : (n - 1); // clamp loads; stores handled below
    int arow = GATHER ? xidx[lrow] : lrow;
    const float* Arow = A + (size_t)arow * 16;

    v8f c = {0.f, 0.f, 0.f, 0.f, 0.f, 0.f, 0.f, 0.f};
#pragma unroll
    for (int kc = 0; kc < 16; kc += 4) {
        const int k0 = kc + 2 * g;
        v2f a;
        a.x = Arow[k0];
        a.y = Arow[k0 + 1];
        if (RELU) { a.x = fmaxf(a.x, 0.0f); a.y = fmaxf(a.y, 0.0f); }
        v2f b;
        b.x = W[(k0    ) * 16 + col];
        b.y = W[(k0 + 1) * 16 + col];
        // 8 args: (neg_a, A, neg_b, B, c_mod, C, reuse_a, reuse_b)
        c = __builtin_amdgcn_wmma_f32_16x16x4_f32(
                false, a, false, b, (short)0, c, false, false);
    }

    float* obase = out + (size_t)tile * 16 * 16 + (size_t)(8 * g) * 16 + col;
    if (tile * 16 + 16 <= n) {
        // full tile (the common case): 8 straight stores, no exec juggling
#pragma unroll
        for (int v = 0; v < 8; ++v) obase[v * 16] = c[v];
    } else {
        // ragged last tile only
#pragma unroll
        for (int v = 0; v < 8; ++v) {
            int orow = tile * 16 + v + 8 * g;
            if (orow < n) obase[v * 16] = c[v];
        }
    }
}

// ---------------------------------------------------------------------------
// out[i][f] = bias[f] + m[i][f] * dis[i]^2   (bias + self-loop message)
// ---------------------------------------------------------------------------
__global__ __launch_bounds__(TPB)
void k_init_out(const float* __restrict__ m, const float* __restrict__ dis,
                const float* __restrict__ bias, float* __restrict__ out, int n) {
    int i = blockIdx.x * TPB + threadIdx.x;
    if (i >= n) return;
    float w = dis[i]; w = w * w;
    const float4* mr = (const float4*)(m + (size_t)i * 16);
    const float4* bb = (const float4*)bias;
    float4*       orow = (float4*)(out + (size_t)i * 16);
#pragma unroll
    for (int q = 0; q < 4; ++q) {
        float4 mv = mr[q], bv = bb[q], r;
        r.x = bv.x + mv.x * w; r.y = bv.y + mv.y * w;
        r.z = bv.z + mv.z * w; r.w = bv.w + mv.w * w;
        orow[q] = r;
    }
}

// ---------------------------------------------------------------------------
// Edge scatter: out[dst] += m[src] * (dis[src]*dis[dst])  (16 f32 atomics/edge)
// Working set (~98MB) is L2-resident (192MB), so gathers + atomics resolve in GL2.
// ---------------------------------------------------------------------------
__global__ __launch_bounds__(TPB)
void k_scatter(const int* __restrict__ ei, const float* __restrict__ m,
               const float* __restrict__ dis, float* __restrict__ out,
               int e) {
    int i = blockIdx.x * TPB + threadIdx.x;
    if (i >= e) return;
    int s = ei[i];
    int d = ei[e + i];
    float w = dis[s] * dis[d];
    const float4* mr = (const float4*)(m + (size_t)s * 16);
    float* orow = out + (size_t)d * 16;
#pragma unroll
    for (int q = 0; q < 4; ++q) {
        float4 v = mr[q];
        atomicAdd(orow + 4 * q + 0, v.x * w);
        atomicAdd(orow + 4 * q + 1, v.y * w);
        atomicAdd(orow + 4 * q + 2, v.z * w);
        atomicAdd(orow + 4 * q + 3, v.w * w);
    }
}

// ---------------------------------------------------------------------------
// Decode: out[i] = fcb + <z[a], fcW[0:16]> + <z[b], fcW[16:32]>
// ---------------------------------------------------------------------------
__global__ __launch_bounds__(TPB)
void k_decode(const int* __restrict__ eli, const float* __restrict__ z,
              const float* __restrict__ fcW, const float* __restrict__ fcb,
              float* __restrict__ out, int nl) {
    int i = blockIdx.x * TPB + threadIdx.x;
    if (i >= nl) return;
    int a = eli[i];
    int b = eli[nl + i];
    const float4* za = (const float4*)(z + (size_t)a * 16);
    const float4* zb = (const float4*)(z + (size_t)b * 16);
    const float4* wa = (const float4*)fcW;
    const float4* wb = (const float4*)(fcW + 16);
    float acc = fcb[0];
#pragma unroll
    for (int q = 0; q < 4; ++q) {
        float4 v = za[q], w = wa[q];
        acc += v.x * w.x + v.y * w.y + v.z * w.z + v.w * w.w;
        v = zb[q]; w = wb[q];
        acc += v.x * w.x + v.y * w.y + v.z * w.z + v.w * w.w;
    }
    out[i] = acc;
}

extern "C" void kernel_launch(void* const* d_in, const int* in_sizes, int n_in,
                              void* d_out, int out_size, void* d_ws, size_t ws_size,
                              hipStream_t stream) {
    (void)n_in; (void)out_size; (void)ws_size;
    const int*   x    = (const int*)d_in[0];
    const int*   ei   = (const int*)d_in[1];   // [2, E] flat: row0=src, row1=dst
    const int*   eli  = (const int*)d_in[2];   // [2, NL] flat
    const float* emb  = (const float*)d_in[3]; // [N, 16]
    const float* W1   = (const float*)d_in[4]; // [16, 16]
    const float* b1   = (const float*)d_in[5]; // [16]
    const float* W2   = (const float*)d_in[6]; // [16, 16]
    const float* b2   = (const float*)d_in[7]; // [16]
    const float* fcW  = (const float*)d_in[8]; // [32, 1]
    const float* fcb  = (const float*)d_in[9]; // [1]

    const int N  = in_sizes[3] / 16;
    const int E  = in_sizes[1] / 2;
    const int NL = in_sizes[2] / 2;

    // Workspace layout (floats): dis[N] | m[N*16] | o1[N*16] | z[N*16] ~= 98MB
    float* dis = (float*)d_ws;
    float* m   = dis + (size_t)N;          // GEMM output, reused for both layers
    float* o1  = m   + (size_t)N * 16;     // layer-1 aggregation output
    float* z   = o1  + (size_t)N * 16;     // layer-2 aggregation output

    const dim3 gN((N  + TPB - 1) / TPB);
    const dim3 gE((E  + TPB - 1) / TPB);
    const dim3 gL((NL + TPB - 1) / TPB);
    const int  ntiles = (N + 15) / 16;
    const dim3 gT((ntiles + WAVES_PER_BLOCK - 1) / WAVES_PER_BLOCK);

    // degree -> dis = rsqrt(deg) (self-loops folded in as init=1)
    k_fill_deg <<<gN, TPB, 0, stream>>>(dis, N);
    k_deg_edges<<<gE, TPB, 0, stream>>>(ei + E, dis, E);
    k_dis      <<<gN, TPB, 0, stream>>>(dis, N);

    // layer 1: m = emb[x] @ W1 ; o1 = b1 + selfloop + scatter(edges)
    k_gemm16_wmma<true,  false><<<gT, TPB, 0, stream>>>(emb, x, W1, m, N);
    k_init_out<<<gN, TPB, 0, stream>>>(m, dis, b1, o1, N);
    k_scatter <<<gE, TPB, 0, stream>>>(ei, m, dis, o1, E);

    // layer 2: m = relu(o1) @ W2 (relu fused into A loads); z = b2 + agg
    k_gemm16_wmma<false, true><<<gT, TPB, 0, stream>>>(o1, nullptr, W2, m, N);
    k_init_out<<<gN, TPB, 0, stream>>>(m, dis, b2, z, N);
    k_scatter <<<gE, TPB, 0, stream>>>(ei, m, dis, z, E);

    // decode
    k_decode<<<gL, TPB, 0, stream>>>(eli, z, fcW, fcb, (float*)d_out, NL);
}